// BidirectionalCrossModalScan_70738111365277
// MI455X (gfx1250) — compile-verified
//
#include <hip/hip_runtime.h>

#define BB 4
#define NSEQ 2048
#define LSEQ 4096            // 2*NSEQ
#define DIMM 192
#define DINN 384
#define NLAYERS 4
#define NST 64
#define QQ 128
#define NCH 32               // LSEQ/QQ
#define HDIM 64
#define HH 6
#define KK 4
#define CONVC 512            // DIN + 2*NSTATE
#define DPROJJ 902           // 2*DIN + 2*NSTATE + H
#define BLROWS (BB*LSEQ)     // 16384

typedef __attribute__((ext_vector_type(2))) float v2f;
typedef __attribute__((ext_vector_type(8))) float v8f;

// ---------------------------------------------------------------------------
// Generic batched GEMM via V_WMMA_F32_16X16X4_F32 (fp32 matrix pipe, wave32).
// One wave per 16x32 output macro-tile (2 accumulators share the A fragment).
// Compile-time TA/TB/ACC/GUARD -> tight branch-free inner loop.
// Assumes M % 16 == 0 and K % 4 == 0 (true for every call site here);
// GUARD adds column guards for ragged N (only the W_in GEMM, N=902).
// ---------------------------------------------------------------------------
struct GemmP {
  const float* A; const float* B; float* C;
  int M, N, K, lda, ldb, ldc;
  long long sAb, sAc, sAh, sBb, sBc, sBh, sCb, sCc, sCh;
  int nC, nH;
};

template<int TA, int TB, int ACC, int GUARD>
__global__ __launch_bounds__(32) void gemm_k(GemmP g) {
  int z = blockIdx.z;
  int h = z % g.nH;
  int c = (z / g.nH) % g.nC;
  int b = z / (g.nH * g.nC);
  const float* __restrict__ A  = g.A + (long long)b * g.sAb + (long long)c * g.sAc + (long long)h * g.sAh;
  const float* __restrict__ Bp = g.B + (long long)b * g.sBb + (long long)c * g.sBc + (long long)h * g.sBh;
  float* __restrict__ C        = g.C + (long long)b * g.sCb + (long long)c * g.sCc + (long long)h * g.sCh;
  const int lda = g.lda, ldb = g.ldb, ldc = g.ldc, K = g.K, N = g.N;

  int row0 = blockIdx.y * 16;
  int col0 = blockIdx.x * 32;
  int lane = threadIdx.x & 31;
  int m    = lane & 15;        // A row within tile / B,D column within tile
  int kp   = lane >> 4;        // K-pair select / D row-half select
  int ra   = row0 + m;
  int cb0  = col0 + m;
  int cb1  = cb0 + 16;
  bool ok0 = !GUARD || (cb0 < N);
  bool ok1 = !GUARD || (cb1 < N);

  v8f acc0 = {0.f,0.f,0.f,0.f,0.f,0.f,0.f,0.f};
  v8f acc1 = {0.f,0.f,0.f,0.f,0.f,0.f,0.f,0.f};
  if (ACC) {
#pragma unroll
    for (int r = 0; r < 8; ++r) {
      int rr = row0 + r + 8 * kp;
      if (ok0) acc0[r] = C[(long long)rr * ldc + cb0];
      if (ok1) acc1[r] = C[(long long)rr * ldc + cb1];
    }
  }

#pragma unroll 4
  for (int k0 = 0; k0 < K; k0 += 4) {
    int ka = k0 + 2 * kp;
    v2f a, b0, b1;
    if (TA) {
      a[0] = A[(long long)ka * lda + ra];
      a[1] = A[(long long)(ka + 1) * lda + ra];
    } else {
      const float* ap = A + (long long)ra * lda + ka;   // consecutive -> b64 load
      a[0] = ap[0];
      a[1] = ap[1];
    }
    if (TB) {
      b0[0] = ok0 ? Bp[(long long)cb0 * ldb + ka    ] : 0.f;   // consecutive pair
      b0[1] = ok0 ? Bp[(long long)cb0 * ldb + ka + 1] : 0.f;
      b1[0] = ok1 ? Bp[(long long)cb1 * ldb + ka    ] : 0.f;
      b1[1] = ok1 ? Bp[(long long)cb1 * ldb + ka + 1] : 0.f;
    } else {
      b0[0] = ok0 ? Bp[(long long)ka * ldb + cb0      ] : 0.f;
      b0[1] = ok0 ? Bp[(long long)(ka + 1) * ldb + cb0] : 0.f;
      b1[0] = ok1 ? Bp[(long long)ka * ldb + cb1      ] : 0.f;
      b1[1] = ok1 ? Bp[(long long)(ka + 1) * ldb + cb1] : 0.f;
    }
    // 8 args: (neg_a, A, neg_b, B, c_mod, C, reuse_a, reuse_b)
    acc0 = __builtin_amdgcn_wmma_f32_16x16x4_f32(false, a, false, b0, (short)0, acc0, false, false);
    acc1 = __builtin_amdgcn_wmma_f32_16x16x4_f32(false, a, false, b1, (short)0, acc1, false, false);
  }

#pragma unroll
  for (int r = 0; r < 8; ++r) {
    int rr = row0 + r + 8 * kp;
    if (ok0) C[(long long)rr * ldc + cb0] = acc0[r];
    if (ok1) C[(long long)rr * ldc + cb1] = acc1[r];
  }
}

// ---------------------------------------------------------------------------
// Elementwise / reduction kernels
// ---------------------------------------------------------------------------
__device__ __forceinline__ float siluf(float x) { return x / (1.f + expf(-x)); }

__global__ void interleave_k(const float* sem, const float* feat, float* ufwd, float* ubwd) {
  long long idx = (long long)blockIdx.x * blockDim.x + threadIdx.x;
  long long total = (long long)BB * LSEQ * DIMM;
  if (idx >= total) return;
  int d = (int)(idx % DIMM);
  long long t = idx / DIMM;
  int l = (int)(t % LSEQ);
  int b = (int)(t / LSEQ);
  int n = l >> 1, which = l & 1;
  float v = (which ? feat : sem)[((long long)b * NSEQ + n) * DIMM + d];
  ufwd[idx] = v;
  ubwd[((long long)b * LSEQ + (LSEQ - 1 - l)) * DIMM + d] = v;
}

// LayerNorm, one wave32 per row (width 192 = 32*6)
__global__ __launch_bounds__(32) void layernorm_k(const float* x, const float* w, const float* bia,
                                                  float* out, int rows, int width) {
  int row = blockIdx.x;
  if (row >= rows) return;
  const float* xr = x + (long long)row * width;
  int lane = threadIdx.x;
  float s = 0.f;
  for (int i = lane; i < width; i += 32) s += xr[i];
  for (int o = 16; o > 0; o >>= 1) s += __shfl_xor(s, o, 32);
  float mean = s / width;
  float v = 0.f;
  for (int i = lane; i < width; i += 32) { float d = xr[i] - mean; v += d * d; }
  for (int o = 16; o > 0; o >>= 1) v += __shfl_xor(v, o, 32);
  float inv = rsqrtf(v / width + 1e-5f);
  float* orow = out + (long long)row * width;
  for (int i = lane; i < width; i += 32) orow[i] = (xr[i] - mean) * inv * w[i] + bia[i];
}

// Depthwise causal conv (K=4, left pad 3) + SiLU on zx[:, DIN:DIN+CONV]
__global__ void conv_silu_k(const float* zx, const float* w, const float* bias, float* xBC) {
  long long idx = (long long)blockIdx.x * blockDim.x + threadIdx.x;
  long long total = (long long)BB * LSEQ * CONVC;
  if (idx >= total) return;
  int ch = (int)(idx % CONVC);
  long long t = idx / CONVC;
  int l = (int)(t % LSEQ);
  int b = (int)(t / LSEQ);
  float a = bias[ch];
#pragma unroll
  for (int k = 0; k < KK; ++k) {
    int ls = l - (KK - 1) + k;
    if (ls >= 0) a += w[ch * KK + k] * zx[((long long)b * LSEQ + ls) * DPROJJ + DINN + ch];
  }
  xBC[idx] = siluf(a);
}

__global__ void dt_k(const float* zx, const float* dt_bias, float* dt) {
  long long idx = (long long)blockIdx.x * blockDim.x + threadIdx.x;
  long long total = (long long)BLROWS * HH;
  if (idx >= total) return;
  int h = (int)(idx % HH);
  long long t = idx / HH;
  float x = zx[t * DPROJJ + DINN + CONVC + h] + dt_bias[h];
  dt[idx] = (x > 20.f) ? x : log1pf(expf(x));
}

// Per-(b,h,chunk) inclusive cumsum of dA = dt*A over the 128 chunk positions
__global__ void cumsum_k(const float* dt, const float* A_log, float* dAcs, float* dAsum) {
  int idx = blockIdx.x * blockDim.x + threadIdx.x;
  if (idx >= BB * HH * NCH) return;
  int c = idx % NCH;
  int h = (idx / NCH) % HH;
  int b = idx / (NCH * HH);
  float A = -expf(A_log[h]);
  float run = 0.f;
  for (int q = 0; q < QQ; ++q) {
    int l = c * QQ + q;
    run += dt[((long long)b * LSEQ + l) * HH + h] * A;
    dAcs[((long long)(b * HH + h)) * LSEQ + l] = run;
  }
  dAsum[idx] = run;
}

// xs = x*dt ;  xdecay = xs * exp(chunk_end_cs - cs[l])
__global__ void xs_k(const float* xBC, const float* dt, const float* dAcs, const float* dAsum,
                     float* xs, float* xdecay) {
  long long idx = (long long)blockIdx.x * blockDim.x + threadIdx.x;
  long long total = (long long)BLROWS * DINN;
  if (idx >= total) return;
  int col = (int)(idx % DINN);
  long long t = idx / DINN;
  int l = (int)(t % LSEQ);
  int b = (int)(t / LSEQ);
  int h = col / HDIM;
  int c = l / QQ;
  float s = xBC[t * CONVC + col] * dt[t * HH + h];
  float decay = expf(dAsum[((long long)b * HH + h) * NCH + c] - dAcs[((long long)b * HH + h) * LSEQ + l]);
  xs[idx] = s;
  xdecay[idx] = s * decay;
}

// Gh[l,s] = (s<=l) ? Gbase[l,s] * exp(cs[l]-cs[s]) : 0      (per b,c,h)
__global__ void gmask_k(const float* Gbase, const float* dAcs, float* G) {
  long long idx = (long long)blockIdx.x * blockDim.x + threadIdx.x;
  long long total = (long long)BB * NCH * HH * QQ * QQ;
  if (idx >= total) return;
  int s = (int)(idx % QQ);
  long long t = idx / QQ;
  int l = (int)(t % QQ); t /= QQ;
  int h = (int)(t % HH); t /= HH;
  int c = (int)(t % NCH);
  int b = (int)(t / NCH);
  float v = 0.f;
  if (s <= l) {
    float g = Gbase[(((long long)(b * NCH + c)) * QQ + l) * QQ + s];
    long long base = ((long long)b * HH + h) * LSEQ + (long long)c * QQ;
    v = g * expf(dAcs[base + l] - dAcs[base + s]);
  }
  G[idx] = v;
}

// Sequential inter-chunk state recurrence: prevS[c] = S ; S = exp(dAsum[c])*S + states[c]
__global__ void chunkscan_k(const float* states, const float* dAsum, float* prevS) {
  int idx = blockIdx.x * blockDim.x + threadIdx.x;
  int total = BB * HH * HDIM * NST;
  if (idx >= total) return;
  int n = idx % NST;
  int p = (idx / NST) % HDIM;
  int h = (idx / (NST * HDIM)) % HH;
  int b = idx / (NST * HDIM * HH);
  float S = 0.f;
  for (int c = 0; c < NCH; ++c) {
    long long e = (((long long)(b * NCH + c)) * HH + h) * (HDIM * NST) + p * NST + n;
    prevS[e] = S;
    S = expf(dAsum[((long long)b * HH + h) * NCH + c]) * S + states[e];
  }
}

// Cs[b,h,l,n] = C[l,n] * exp(cs[b,h,l])
__global__ void cscale_k(const float* xBC, const float* dAcs, float* Cs) {
  long long idx = (long long)blockIdx.x * blockDim.x + threadIdx.x;
  long long total = (long long)BB * HH * LSEQ * NST;
  if (idx >= total) return;
  int n = (int)(idx % NST);
  long long t = idx / NST;
  int l = (int)(t % LSEQ); t /= LSEQ;
  int h = (int)(t % HH);
  int b = (int)(t / HH);
  float cv = xBC[((long long)b * LSEQ + l) * CONVC + DINN + NST + n];
  Cs[idx] = cv * expf(dAcs[((long long)b * HH + h) * LSEQ + l]);
}

// Y += D[h] * x (unscaled conv output)
__global__ void addDx_k(const float* xBC, const float* Dp, float* Y) {
  long long idx = (long long)blockIdx.x * blockDim.x + threadIdx.x;
  long long total = (long long)BLROWS * DINN;
  if (idx >= total) return;
  int col = (int)(idx % DINN);
  long long t = idx / DINN;
  Y[idx] += Dp[col / HDIM] * xBC[t * CONVC + col];
}

// yg = rmsnorm(Y * silu(z), w), one wave per 384-wide row (12 elems/lane)
__global__ __launch_bounds__(32) void gate_rms_k(const float* Y, const float* zx, const float* w,
                                                 float* out, int rows) {
  int row = blockIdx.x;
  if (row >= rows) return;
  const float* yr = Y + (long long)row * DINN;
  const float* zr = zx + (long long)row * DPROJJ;  // z = first DIN cols of zxbcdt
  int lane = threadIdx.x;
  float tvals[12];
  float ss = 0.f;
#pragma unroll
  for (int j = 0; j < 12; ++j) {
    int i = lane + 32 * j;
    float val = yr[i] * siluf(zr[i]);
    tvals[j] = val;
    ss += val * val;
  }
  for (int o = 16; o > 0; o >>= 1) ss += __shfl_xor(ss, o, 32);
  float inv = rsqrtf(ss / DINN + 1e-5f);
  float* orow = out + (long long)row * DINN;
#pragma unroll
  for (int j = 0; j < 12; ++j) { int i = lane + 32 * j; orow[i] = tvals[j] * inv * w[i]; }
}

// cat = [fwd, reverse(bwd)]
__global__ void catrev_k(const float* fwd, const float* bwd, float* catb) {
  long long idx = (long long)blockIdx.x * blockDim.x + threadIdx.x;
  long long total = (long long)BLROWS * DIMM;
  if (idx >= total) return;
  int d = (int)(idx % DIMM);
  long long t = idx / DIMM;
  int l = (int)(t % LSEQ);
  int b = (int)(t / LSEQ);
  long long base = t * (2 * DIMM);
  catb[base + d] = fwd[idx];
  catb[base + DIMM + d] = bwd[((long long)b * LSEQ + (LSEQ - 1 - l)) * DIMM + d];
}

__global__ void biasinit_k(const float* bg, float* gate) {
  long long idx = (long long)blockIdx.x * blockDim.x + threadIdx.x;
  long long total = (long long)BLROWS * DIMM;
  if (idx >= total) return;
  gate[idx] = bg[idx % DIMM];
}

// merged = sigmoid(gate)*fwd + (1-sig)*bwd_rev ; de-interleave into the two outputs
__global__ void merge_k(const float* gate, const float* fwd, const float* bwd, float* out) {
  long long idx = (long long)blockIdx.x * blockDim.x + threadIdx.x;
  long long total = (long long)BLROWS * DIMM;
  if (idx >= total) return;
  int d = (int)(idx % DIMM);
  long long t = idx / DIMM;
  int l = (int)(t % LSEQ);
  int b = (int)(t / LSEQ);
  float g = 1.f / (1.f + expf(-gate[idx]));
  float fv = fwd[idx];
  float bv = bwd[((long long)b * LSEQ + (LSEQ - 1 - l)) * DIMM + d];
  float mv = g * fv + (1.f - g) * bv;
  int n = l >> 1, which = l & 1;  // even -> output0 (semantic), odd -> output1 (features)
  out[((long long)which * BB * NSEQ + (long long)b * NSEQ + n) * DIMM + d] = mv;
}

// ---------------------------------------------------------------------------
// Host orchestration
// ---------------------------------------------------------------------------
static inline int cdiv(long long a, int b) { return (int)((a + b - 1) / b); }

static void launch_gemm(hipStream_t st, const float* A, const float* B, float* C,
                        int M, int N, int K, int lda, int ldb, int ldc,
                        long long sAb, long long sAc, long long sAh,
                        long long sBb, long long sBc, long long sBh,
                        long long sCb, long long sCc, long long sCh,
                        int nB, int nC, int nH, int tA, int tB, int accum) {
  GemmP g{A, B, C, M, N, K, lda, ldb, ldc, sAb, sAc, sAh, sBb, sBc, sBh, sCb, sCc, sCh, nC, nH};
  dim3 grid((N + 31) / 32, (M + 15) / 16, nB * nC * nH);
  int guard = (N % 32 != 0) ? 1 : 0;
  int key = (tA << 3) | (tB << 2) | (accum << 1) | guard;
  switch (key) {
    case 0x0: gemm_k<0,0,0,0><<<grid, 32, 0, st>>>(g); break;
    case 0x1: gemm_k<0,0,0,1><<<grid, 32, 0, st>>>(g); break;
    case 0x2: gemm_k<0,0,1,0><<<grid, 32, 0, st>>>(g); break;
    case 0x3: gemm_k<0,0,1,1><<<grid, 32, 0, st>>>(g); break;
    case 0x4: gemm_k<0,1,0,0><<<grid, 32, 0, st>>>(g); break;
    case 0x5: gemm_k<0,1,0,1><<<grid, 32, 0, st>>>(g); break;
    case 0x6: gemm_k<0,1,1,0><<<grid, 32, 0, st>>>(g); break;
    case 0x7: gemm_k<0,1,1,1><<<grid, 32, 0, st>>>(g); break;
    case 0x8: gemm_k<1,0,0,0><<<grid, 32, 0, st>>>(g); break;
    case 0x9: gemm_k<1,0,0,1><<<grid, 32, 0, st>>>(g); break;
    case 0xA: gemm_k<1,0,1,0><<<grid, 32, 0, st>>>(g); break;
    case 0xB: gemm_k<1,0,1,1><<<grid, 32, 0, st>>>(g); break;
    case 0xC: gemm_k<1,1,0,0><<<grid, 32, 0, st>>>(g); break;
    case 0xD: gemm_k<1,1,0,1><<<grid, 32, 0, st>>>(g); break;
    case 0xE: gemm_k<1,1,1,0><<<grid, 32, 0, st>>>(g); break;
    case 0xF: gemm_k<1,1,1,1><<<grid, 32, 0, st>>>(g); break;
  }
}

struct Params {
  const float *ln_w, *ln_b, *W_in, *conv_w, *conv_b, *dt_bias, *A_log, *D, *norm_w, *W_out, *fln_w, *fln_b;
};

extern "C" void kernel_launch(void* const* d_in, const int* in_sizes, int n_in,
                              void* d_out, int out_size, void* d_ws, size_t ws_size,
                              hipStream_t stream) {
  (void)in_sizes; (void)n_in; (void)out_size; (void)ws_size;
  // setup_inputs() dict order: semantic, features, fwd_params(12 tensors), bwd_params(12), Wg, bg
  const float* sem  = (const float*)d_in[0];
  const float* feat = (const float*)d_in[1];
  auto getP = [&](int base) {
    Params p;
    p.ln_w    = (const float*)d_in[base + 0];
    p.ln_b    = (const float*)d_in[base + 1];
    p.W_in    = (const float*)d_in[base + 2];
    p.conv_w  = (const float*)d_in[base + 3];
    p.conv_b  = (const float*)d_in[base + 4];
    p.dt_bias = (const float*)d_in[base + 5];
    p.A_log   = (const float*)d_in[base + 6];
    p.D       = (const float*)d_in[base + 7];
    p.norm_w  = (const float*)d_in[base + 8];
    p.W_out   = (const float*)d_in[base + 9];
    p.fln_w   = (const float*)d_in[base + 10];
    p.fln_b   = (const float*)d_in[base + 11];
    return p;
  };
  Params pf = getP(2), pb = getP(14);
  const float* Wg = (const float*)d_in[26];
  const float* bg = (const float*)d_in[27];
  float* outp = (float*)d_out;

  // ---- workspace bump allocation (floats) ----
  float* ws = (float*)d_ws;
  size_t off = 0;
  auto alloc = [&](size_t n) { float* p = ws + off; off += n; return p; };
  float* u_fwd   = alloc((size_t)BLROWS * DIMM);
  float* u_bwd   = alloc((size_t)BLROWS * DIMM);
  float* normed  = alloc((size_t)BLROWS * DIMM);
  float* zx      = alloc((size_t)BLROWS * DPROJJ);
  float* xBC     = alloc((size_t)BLROWS * CONVC);
  float* dtb     = alloc((size_t)BLROWS * HH);
  float* dAcs    = alloc((size_t)BB * HH * LSEQ);
  float* dAsum   = alloc((size_t)BB * HH * NCH);
  float* xs      = alloc((size_t)BLROWS * DINN);
  float* xdecay  = alloc((size_t)BLROWS * DINN);    // reused as Cs (after states GEMM)
  float* Gbase   = alloc((size_t)BB * NCH * QQ * QQ);
  float* G       = alloc((size_t)BB * NCH * HH * QQ * QQ);
  float* states  = alloc((size_t)BB * NCH * HH * HDIM * NST);
  float* prevS   = alloc((size_t)BB * NCH * HH * HDIM * NST);
  float* Y       = alloc((size_t)BLROWS * DINN);
  float* ygated  = alloc((size_t)BLROWS * DINN);
  float* out_fwd = alloc((size_t)BLROWS * DIMM);
  float* out_bwd = alloc((size_t)BLROWS * DIMM);
  float* Cs   = xdecay;           // same size, non-overlapping lifetime within a layer
  float* catb = Gbase;            // finale only: Gbase+G region (14.6M floats contiguous)
  float* gate = Gbase + (size_t)BLROWS * DINN;

  const int TPB = 256;
  long long nUD = (long long)BLROWS * DIMM;

  hipLaunchKernelGGL(interleave_k, dim3(cdiv(nUD, TPB)), dim3(TPB), 0, stream, sem, feat, u_fwd, u_bwd);

  auto run_stack = [&](float* u, const Params& p, float* outbuf) {
    for (int i = 0; i < NLAYERS; ++i) {
      // 1. pre-LN
      hipLaunchKernelGGL(layernorm_k, dim3(BLROWS), dim3(32), 0, stream,
                         u, p.ln_w + i * DIMM, p.ln_b + i * DIMM, normed, BLROWS, DIMM);
      // 2. zxbcdt = normed @ W_in   (16384 x 902 x 192, ragged N -> GUARD)
      launch_gemm(stream, normed, p.W_in + (long long)i * DIMM * DPROJJ, zx,
                  BLROWS, DPROJJ, DIMM, DIMM, DPROJJ, DPROJJ,
                  0,0,0, 0,0,0, 0,0,0, 1,1,1, 0,0,0);
      // 3. causal conv + silu -> xBC
      hipLaunchKernelGGL(conv_silu_k, dim3(cdiv((long long)BLROWS * CONVC, TPB)), dim3(TPB), 0, stream,
                         zx, p.conv_w + i * CONVC * KK, p.conv_b + i * CONVC, xBC);
      // 4. dt = softplus(dt_raw + bias)
      hipLaunchKernelGGL(dt_k, dim3(cdiv((long long)BLROWS * HH, TPB)), dim3(TPB), 0, stream,
                         zx, p.dt_bias + i * HH, dtb);
      // 5. per-chunk cumsum of dA
      hipLaunchKernelGGL(cumsum_k, dim3(cdiv(BB * HH * NCH, TPB)), dim3(TPB), 0, stream,
                         dtb, p.A_log + i * HH, dAcs, dAsum);
      // 6. xs = x*dt, xdecay = xs*exp(end-cs)
      hipLaunchKernelGGL(xs_k, dim3(cdiv((long long)BLROWS * DINN, TPB)), dim3(TPB), 0, stream,
                         xBC, dtb, dAcs, dAsum, xs, xdecay);
      // 7. Gbase[b,c] = C(128x64) @ B^T  (batched over b,c)
      launch_gemm(stream, xBC + DINN + NST, xBC + DINN, Gbase,
                  QQ, QQ, NST, CONVC, CONVC, QQ,
                  (long long)LSEQ * CONVC, (long long)QQ * CONVC, 0,
                  (long long)LSEQ * CONVC, (long long)QQ * CONVC, 0,
                  (long long)NCH * QQ * QQ, (long long)QQ * QQ, 0,
                  BB, NCH, 1, 0, 1, 0);
      // 8. per-head causal decay mask
      hipLaunchKernelGGL(gmask_k, dim3(cdiv((long long)BB * NCH * HH * QQ * QQ, TPB)), dim3(TPB), 0, stream,
                         Gbase, dAcs, G);
      // 9. Y_diag = Gh(128x128) @ xs_head(128x64)  (batched b,c,h)
      launch_gemm(stream, G, xs, Y,
                  QQ, HDIM, QQ, QQ, DINN, DINN,
                  (long long)NCH * HH * QQ * QQ, (long long)HH * QQ * QQ, (long long)QQ * QQ,
                  (long long)LSEQ * DINN, (long long)QQ * DINN, HDIM,
                  (long long)LSEQ * DINN, (long long)QQ * DINN, HDIM,
                  BB, NCH, HH, 0, 0, 0);
      // 10. states = xdecay^T(64x128) @ B(128x64)
      launch_gemm(stream, xdecay, xBC + DINN, states,
                  HDIM, NST, QQ, DINN, CONVC, NST,
                  (long long)LSEQ * DINN, (long long)QQ * DINN, HDIM,
                  (long long)LSEQ * CONVC, (long long)QQ * CONVC, 0,
                  (long long)NCH * HH * HDIM * NST, (long long)HH * HDIM * NST, (long long)HDIM * NST,
                  BB, NCH, HH, 1, 0, 0);
      // 11. inter-chunk recurrence
      hipLaunchKernelGGL(chunkscan_k, dim3(cdiv(BB * HH * HDIM * NST, TPB)), dim3(TPB), 0, stream,
                         states, dAsum, prevS);
      // 12. Cs = C * exp(cs)   (xdecay lifetime over -> alias is safe)
      hipLaunchKernelGGL(cscale_k, dim3(cdiv((long long)BB * HH * LSEQ * NST, TPB)), dim3(TPB), 0, stream,
                         xBC, dAcs, Cs);
      // 13. Y += Cs(128x64) @ prevS^T(64x64)
      launch_gemm(stream, Cs, prevS, Y,
                  QQ, HDIM, NST, NST, NST, DINN,
                  (long long)HH * LSEQ * NST, (long long)QQ * NST, (long long)LSEQ * NST,
                  (long long)NCH * HH * HDIM * NST, (long long)HH * HDIM * NST, (long long)HDIM * NST,
                  (long long)LSEQ * DINN, (long long)QQ * DINN, HDIM,
                  BB, NCH, HH, 0, 1, 1);
      // 14. Y += D[h]*x
      hipLaunchKernelGGL(addDx_k, dim3(cdiv((long long)BLROWS * DINN, TPB)), dim3(TPB), 0, stream,
                         xBC, p.D + i * HH, Y);
      // 15. ygated = rmsnorm(Y * silu(z), norm_w)
      hipLaunchKernelGGL(gate_rms_k, dim3(BLROWS), dim3(32), 0, stream,
                         Y, zx, p.norm_w + i * DINN, ygated, BLROWS);
      // 16. u += ygated @ W_out   (residual, in-place accumulate)
      launch_gemm(stream, ygated, p.W_out + (long long)i * DINN * DIMM, u,
                  BLROWS, DIMM, DINN, DINN, DIMM, DIMM,
                  0,0,0, 0,0,0, 0,0,0, 1,1,1, 0,0,1);
    }
    hipLaunchKernelGGL(layernorm_k, dim3(BLROWS), dim3(32), 0, stream,
                       u, p.fln_w, p.fln_b, outbuf, BLROWS, DIMM);
  };

  run_stack(u_fwd, pf, out_fwd);
  run_stack(u_bwd, pb, out_bwd);

  // gating: gate = sigmoid([fwd, rev(bwd)] @ Wg + bg)
  hipLaunchKernelGGL(catrev_k, dim3(cdiv(nUD, TPB)), dim3(TPB), 0, stream, out_fwd, out_bwd, catb);
  hipLaunchKernelGGL(biasinit_k, dim3(cdiv(nUD, TPB)), dim3(TPB), 0, stream, bg, gate);
  launch_gemm(stream, catb, Wg, gate,
              BLROWS, DIMM, 2 * DIMM, 2 * DIMM, DIMM, DIMM,
              0,0,0, 0,0,0, 0,0,0, 1,1,1, 0,0,1);
  hipLaunchKernelGGL(merge_k, dim3(cdiv(nUD, TPB)), dim3(TPB), 0, stream, gate, out_fwd, out_bwd, outp);
}